// OAMPNet_52295521796665
// MI455X (gfx1250) — compile-verified
//
#include <hip/hip_runtime.h>
#include <math.h>

// OAMP-Net MIMO detector for MI455X (gfx1250), fp32 end-to-end.
// One workgroup (256 threads = 8 wave32) per batch element.
// All 64x64x64 matmuls use V_WMMA_F32_16X16X4_F32 (native fp32 WMMA).

typedef __attribute__((ext_vector_type(2))) float v2f;
typedef __attribute__((ext_vector_type(8))) float v8f;

static constexpr int LSTR = 65;   // padded LDS row stride (65 % 64 == 1 -> conflict-free columns)
static constexpr int N2   = 64;   // 2N == 2M
static constexpr int NIT  = 10;

struct LDSBlk {
  float Hs[N2 * LSTR];    // H (row-major)
  float HHTs[N2 * LSTR];  // H H^T
  float Wk[N2 * LSTR];    // workspace: A -> A^-1 -> W
  float red[256];         // reduction scratch
  float ys[N2];
  float xv[N2];
  float zs[N2];
};

__device__ __forceinline__ float block_reduce_sum(float v, float* red) {
  const int tid = threadIdx.x;
  red[tid] = v;
  __syncthreads();
  if (tid == 0) {
    float s = 0.0f;
    #pragma unroll 8
    for (int i = 0; i < 256; ++i) s += red[i];
    red[0] = s;
  }
  __syncthreads();
  const float r = red[0];
  __syncthreads();
  return r;
}

// 16x16 tile of C = opA(A) * opB(B), 64-long K dim, via 16 chained fp32 WMMAs.
// A-fragment (16x4): lanes 0..15 hold K={k0,k0+1}, lanes 16..31 hold K={k0+2,k0+3}.
template <bool TA, bool TB>
__device__ __forceinline__ v8f tile_mm64(const float* __restrict__ A,
                                         const float* __restrict__ B,
                                         int row0, int col0) {
  const int lane = threadIdx.x & 31;
  const int m  = lane & 15;          // row within tile (A) / col within tile (B)
  const int kk = (lane >> 4) << 1;   // K sub-offset per half-wave
  v8f c = {};
  #pragma unroll
  for (int k0 = 0; k0 < 64; k0 += 4) {
    const int ka = k0 + kk;
    v2f a, b;
    if (!TA) { a[0] = A[(row0 + m) * LSTR + ka];     a[1] = A[(row0 + m) * LSTR + ka + 1]; }
    else     { a[0] = A[ka * LSTR + (row0 + m)];     a[1] = A[(ka + 1) * LSTR + (row0 + m)]; }
    if (!TB) { b[0] = B[ka * LSTR + (col0 + m)];     b[1] = B[(ka + 1) * LSTR + (col0 + m)]; }
    else     { b[0] = B[(col0 + m) * LSTR + ka];     b[1] = B[(col0 + m) * LSTR + ka + 1]; }
    c = __builtin_amdgcn_wmma_f32_16x16x4_f32(false, a, false, b, (short)0, c, false, false);
  }
  return c;
}

// Store a 16x16 fp32 C/D fragment (8 VGPRs: lanes 0..15 -> M=v, lanes 16..31 -> M=8+v).
__device__ __forceinline__ void tile_store(float* __restrict__ C, int row0, int col0,
                                           v8f c, float scale) {
  const int lane = threadIdx.x & 31;
  const int n  = lane & 15;
  const int mb = (lane >> 4) << 3;
  #pragma unroll
  for (int v = 0; v < 8; ++v)
    C[(row0 + mb + v) * LSTR + (col0 + n)] = c[v] * scale;
}

// W = (Nf / tr(W'H)) * W',  W' = v2 * H^T * inv(v2*HHT + sig2h*I).   Result in L.Wk.
__device__ void update_W(float v2, float sig2h, LDSBlk& L) {
  const int tid = threadIdx.x;
  // Build A = v2*HHT + sig2h*I into Wk
  for (int idx = tid; idx < N2 * N2; idx += 256) {
    const int i = idx >> 6, j = idx & 63;
    L.Wk[i * LSTR + j] = v2 * L.HHTs[i * LSTR + j] + ((i == j) ? sig2h : 0.0f);
  }
  __syncthreads();
  // In-place Gauss-Jordan inversion (A is SPD -> no pivoting).
  const int gi = tid >> 2;          // row, 4 threads per row
  const int gj = (tid & 3) << 4;    // 16-column slab
  for (int k = 0; k < N2; ++k) {
    const float d    = L.Wk[k * LSTR + k];
    const float invd = 1.0f / d;
    __syncthreads();
    if (tid < N2)
      L.Wk[k * LSTR + tid] = (tid == k) ? invd : L.Wk[k * LSTR + tid] * invd;
    __syncthreads();
    const float f = L.Wk[gi * LSTR + k];
    __syncthreads();
    if (gi != k) {
      #pragma unroll
      for (int jj = 0; jj < 16; ++jj) {
        const int j = gj + jj;
        const float rk  = L.Wk[k * LSTR + j];
        const float cur = L.Wk[gi * LSTR + j];
        L.Wk[gi * LSTR + j] = (j == k) ? (-f * invd) : (cur - f * rk);
      }
    }
    __syncthreads();
  }
  // W' = v2 * H^T * Ainv via WMMA (compute in regs, then overwrite Wk)
  const int wave = tid >> 5;
  const int t0 = wave, t1 = wave + 8;
  const int r0 = (t0 >> 2) << 4, c0 = (t0 & 3) << 4;
  const int r1 = (t1 >> 2) << 4, c1 = (t1 & 3) << 4;
  v8f acc0 = tile_mm64<true, false>(L.Hs, L.Wk, r0, c0);
  v8f acc1 = tile_mm64<true, false>(L.Hs, L.Wk, r1, c1);
  __syncthreads();                         // all Ainv reads done before overwrite
  tile_store(L.Wk, r0, c0, acc0, v2);
  tile_store(L.Wk, r1, c1, acc1, v2);
  __syncthreads();
  // Normalize: W *= Nf / trace(W H)
  float p = 0.0f;
  for (int idx = tid; idx < N2 * N2; idx += 256) {
    const int n = idx >> 6, m = idx & 63;
    p += L.Wk[n * LSTR + m] * L.Hs[m * LSTR + n];
  }
  const float tr = block_reduce_sum(p, L.red);
  const float sc = 32.0f / tr;             // Nf = 32
  for (int idx = tid; idx < N2 * N2; idx += 256) {
    const int i = idx >> 6, j = idx & 63;
    L.Wk[i * LSTR + j] *= sc;
  }
  __syncthreads();
}

__global__ __launch_bounds__(256)
void oamp_kernel(const float* __restrict__ Hg, const float* __restrict__ yg,
                 const float* __restrict__ sigma2g, const float* __restrict__ gammag,
                 const float* __restrict__ thetag, float* __restrict__ out, int Bn) {
  __shared__ LDSBlk L;
  const int b   = blockIdx.x;
  const int tid = threadIdx.x;
  const float* Hb = Hg + (size_t)b * N2 * N2;

  // Load H -> LDS, accumulate ||H||_F^2 (== tr(H^T H))
  float ss = 0.0f;
  for (int idx = tid; idx < N2 * N2; idx += 256) {
    const float h = Hb[idx];
    const int i = idx >> 6, j = idx & 63;
    L.Hs[i * LSTR + j] = h;
    ss += h * h;
  }
  if (tid < N2) { L.ys[tid] = yg[(size_t)b * N2 + tid]; L.xv[tid] = 0.0f; }
  __syncthreads();

  // HHT = H * H^T via WMMA
  const int wave = tid >> 5;
  const int t0 = wave, t1 = wave + 8;
  const int r0 = (t0 >> 2) << 4, c0 = (t0 & 3) << 4;
  const int r1 = (t1 >> 2) << 4, c1 = (t1 & 3) << 4;
  {
    v8f a0 = tile_mm64<false, true>(L.Hs, L.Hs, r0, c0);
    v8f a1 = tile_mm64<false, true>(L.Hs, L.Hs, r1, c1);
    tile_store(L.HHTs, r0, c0, a0, 1.0f);
    tile_store(L.HHTs, r1, c1, a1, 1.0f);
  }
  const float trHTH = block_reduce_sum(ss, L.red);   // barrier also covers HHT stores
  const float sig2h = sigma2g[b] * 0.5f;

  // v2 init from z = y
  const float yn = (tid < N2) ? L.ys[tid] * L.ys[tid] : 0.0f;
  float v2 = fmaxf((block_reduce_sum(yn, L.red) - 32.0f * sig2h) / trHTH, 1e-6f);

  update_W(v2, sig2h, L);

  // trWW = sum(W*W)
  float pw = 0.0f;
  for (int idx = tid; idx < N2 * N2; idx += 256) {
    const int i = idx >> 6, j = idx & 63;
    const float w = L.Wk[i * LSTR + j];
    pw += w * w;
  }
  const float trWW = block_reduce_sum(pw, L.red);

  // trBB = ||I - theta0*W*H||_F^2 = N2 - 2*th0*tr(WH) + th0^2*||WH||_F^2  (WH never stored)
  const float th0 = thetag[0];
  float sq = 0.0f, trd = 0.0f;
  {
    v8f w0 = tile_mm64<false, false>(L.Wk, L.Hs, r0, c0);
    v8f w1 = tile_mm64<false, false>(L.Wk, L.Hs, r1, c1);
    const int lane = tid & 31;
    const int n  = lane & 15;
    const int mb = (lane >> 4) << 3;
    #pragma unroll
    for (int v = 0; v < 8; ++v) {
      const float e0 = w0[v], e1 = w1[v];
      sq += e0 * e0 + e1 * e1;
      if (r0 + mb + v == c0 + n) trd += e0;
      if (r1 + mb + v == c1 + n) trd += e1;
    }
  }
  const float sumSq = block_reduce_sum(sq, L.red);
  const float trWH  = block_reduce_sum(trd, L.red);
  const float trBB  = 64.0f - 2.0f * th0 * trWH + th0 * th0 * sumSq;

  const size_t xs_tot = (size_t)NIT * Bn * N2;
  const float C0 = -0.70710678118654752440f;
  const float C1 =  0.70710678118654752440f;

  #pragma unroll 1
  for (int it = 0; it < NIT; ++it) {
    const float g  = gammag[it];
    const float th = thetag[it];
    // z = y - H x
    if (tid < N2) {
      float acc = L.ys[tid];
      #pragma unroll 8
      for (int j = 0; j < N2; ++j) acc -= L.Hs[tid * LSTR + j] * L.xv[j];
      L.zs[tid] = acc;
    }
    __syncthreads();
    // r = x + g * (W z)   (kept in a register: per-thread private)
    float r = 0.0f;
    if (tid < N2) {
      float acc = 0.0f;
      #pragma unroll 8
      for (int j = 0; j < N2; ++j) acc += L.Wk[tid * LSTR + j] * L.zs[j];
      r = L.xv[tid] + g * acc;
    }
    // v2 and tau2
    const float zn = (tid < N2) ? L.zs[tid] * L.zs[tid] : 0.0f;
    const float znorm = block_reduce_sum(zn, L.red);
    v2 = fmaxf((znorm - 32.0f * sig2h) / trHTH, 1e-6f);
    const float tau2 = (trBB * v2 + th * th * trWW * sig2h) * (1.0f / 64.0f); // /(2*Nf)
    // QPSK Bayes shrinkage + output
    if (tid < N2) {
      const float inv2t = 0.5f / tau2;
      const float d0 = C0 - r, d1 = C1 - r;
      float p0 = fmaxf(expf(-d0 * d0 * inv2t), 5e-13f);
      float p1 = fmaxf(expf(-d1 * d1 * inv2t), 5e-13f);
      const float s = p0 + p1;
      p0 /= s; p1 /= s;
      const float xn = C0 * p0 + C1 * p1;
      L.xv[tid] = xn;
      out[(size_t)it * Bn * N2 + (size_t)b * N2 + tid] = xn;
      const size_t pb = xs_tot + (((size_t)it * Bn + b) * N2 + tid) * 2;
      out[pb]     = p0;
      out[pb + 1] = p1;
    }
    __syncthreads();
    // New W from updated v2 (safe: old W fully consumed above)
    update_W(v2, sig2h, L);
  }
}

extern "C" void kernel_launch(void* const* d_in, const int* in_sizes, int n_in,
                              void* d_out, int out_size, void* d_ws, size_t ws_size,
                              hipStream_t stream) {
  const float* H      = (const float*)d_in[0];
  const float* y      = (const float*)d_in[1];
  const float* sigma2 = (const float*)d_in[2];
  const float* gamma  = (const float*)d_in[3];
  const float* theta  = (const float*)d_in[4];
  const int B = in_sizes[2];   // sigma2 has one element per batch
  oamp_kernel<<<B, 256, 0, stream>>>(H, y, sigma2, gamma, theta, (float*)d_out, B);
}